// DSnetwork_12352325943915
// MI455X (gfx1250) — compile-verified
//
#include <hip/hip_runtime.h>
#include <math.h>

#define D       128
#define NSUB    262144
#define G       8192
#define LNUM    3
#define NTASKS  10

typedef __attribute__((ext_vector_type(2))) float v2f;
typedef __attribute__((ext_vector_type(8))) float v8f;

// XOR-swizzled transposed-W LDS layout: element (c = out column, k) lives at
// WT[c*128 + (k ^ ((c&15)<<2))].  64 KB exactly; B-fragment reads are one
// aligned contiguous ds_load_b64 per (nt,k) and bank-conflict-free across the
// wave; staging writes are coalesced global reads with only 2-way LDS conflict.
#define WTIDX(c, k) ((c) * D + ((k) ^ (((c) & 15) << 2)))

// -------------------------------------------------------------------------
// Kernel 1: segment starts via lower_bound binary search (idx is sorted).
// seg_start[t] = first i with idx[i] >= t, for t in [0, G]; seg_start[G]=N.
// -------------------------------------------------------------------------
__global__ void seg_start_kernel(const int* __restrict__ idx,
                                 int* __restrict__ seg_start, int n, int g)
{
    int t = blockIdx.x * blockDim.x + threadIdx.x;
    if (t > g) return;
    int lo = 0, hi = n;
    while (lo < hi) {
        int mid = (lo + hi) >> 1;
        if (idx[mid] < t) lo = mid + 1; else hi = mid;
    }
    seg_start[t] = lo;
}

// -------------------------------------------------------------------------
// Kernel 2: segment mean. One block per graph, 128 threads = one channel
// each; rows of a segment are contiguous (sorted idx) -> coalesced loads.
// 4 independent accumulators keep 4 loads in flight per thread.
// -------------------------------------------------------------------------
__global__ void seg_mean_kernel(const float* __restrict__ h,
                                const int* __restrict__ seg_start,
                                float* __restrict__ gmean)
{
    int gidx = blockIdx.x;
    int c    = threadIdx.x;            // 0..127
    int s = seg_start[gidx], e = seg_start[gidx + 1];
    float a0 = 0.f, a1 = 0.f, a2 = 0.f, a3 = 0.f;
    int r = s;
    for (; r + 3 < e; r += 4) {
        a0 += h[(size_t)(r + 0) * D + c];
        a1 += h[(size_t)(r + 1) * D + c];
        a2 += h[(size_t)(r + 2) * D + c];
        a3 += h[(size_t)(r + 3) * D + c];
    }
    for (; r < e; ++r) a0 += h[(size_t)r * D + c];
    float acc = (a0 + a1) + (a2 + a3);
    float cnt = (float)(e - s);
    gmean[(size_t)gidx * D + c] = acc / fmaxf(cnt, 1.0f);
}

// -------------------------------------------------------------------------
// WMMA fp32 GEMM core. Block = 256 threads = 8 waves; each wave computes a
// 16x128 output strip via 8 accumulator tiles of V_WMMA_F32_16X16X4_F32.
// A fragment (16x4 f32): lane l -> row (l&15), k-pair base (l>>4)*2.
// B fragment (4x16 f32): lane l -> col (l&15), k-pair base (l>>4)*2.
// A strip is preloaded in 4 chunks of 8 x v2f with one-chunk-ahead
// double-buffering: global latency hides behind 64 WMMAs while keeping
// live A registers at ~32 (frees VGPRs for B-fragment pipelining).
// -------------------------------------------------------------------------
__device__ __forceinline__ void stage_WT(const float* __restrict__ W, int ldw,
                                         int colOff, float* WT)
{
    for (int i = threadIdx.x; i < D * D; i += blockDim.x) {
        int k = i >> 7, c = i & (D - 1);     // coalesced global read along c
        WT[WTIDX(c, k)] = W[(size_t)k * ldw + colOff + c];
    }
    __syncthreads();
}

__device__ __forceinline__ void gemm_acc_wave(const float* __restrict__ A,
                                              int rowBase, const float* WT,
                                              v8f acc[8])
{
    int lane = threadIdx.x & 31;
    int nlo  = lane & 15;
    int kb2  = (lane >> 4) << 1;                 // 0 or 2
    int xm   = nlo << 2;                         // per-lane xor swizzle mask

    const float* arow  = A + (size_t)(rowBase + nlo) * D + kb2;
    const float* wbase = WT + nlo * D;

    v2f abuf[2][8];
    #pragma unroll
    for (int j = 0; j < 8; ++j)
        abuf[0][j] = *reinterpret_cast<const v2f*>(arow + 4 * j);

    #pragma unroll
    for (int c = 0; c < 4; ++c) {
        if (c + 1 < 4) {                         // prefetch next A chunk
            #pragma unroll
            for (int j = 0; j < 8; ++j)
                abuf[(c + 1) & 1][j] =
                    *reinterpret_cast<const v2f*>(arow + 4 * (8 * (c + 1) + j));
        }
        #pragma unroll
        for (int j = 0; j < 8; ++j) {
            int k = 4 * (8 * c + j) + kb2;
            #pragma unroll
            for (int nt = 0; nt < 8; ++nt) {
                v2f b = *reinterpret_cast<const v2f*>(
                    wbase + nt * 16 * D + (k ^ xm));     // one ds_load_b64
                acc[nt] = __builtin_amdgcn_wmma_f32_16x16x4_f32(
                    false, abuf[c & 1][j], false, b, (short)0, acc[nt],
                    false, false);
            }
        }
    }
}

// -------------------------------------------------------------------------
// Kernel 3: fused layer  h_out = elu(h_in @ W + b + x2[idx[row]]).
// In-place safe: each block reads its rows (into registers) before writing.
// -------------------------------------------------------------------------
__global__ __launch_bounds__(256)
void layer_fused_kernel(const float* __restrict__ hin,
                        const float* __restrict__ Wfc,
                        const float* __restrict__ bfc,
                        const float* __restrict__ x2,
                        const int* __restrict__ idx,
                        float* __restrict__ hout)
{
    __shared__ float WT[D * D];                  // 64 KB, swizzled transpose
    stage_WT(Wfc, D, 0, WT);

    int wave = threadIdx.x >> 5;
    int lane = threadIdx.x & 31;
    int nlo  = lane & 15;
    int hi   = lane >> 4;
    int rowBase = blockIdx.x * 128 + wave * 16;

    v8f acc[8] = {};
    gemm_acc_wave(hin, rowBase, WT, acc);

    float bn[8];
    #pragma unroll
    for (int nt = 0; nt < 8; ++nt) bn[nt] = bfc[nt * 16 + nlo];

    #pragma unroll
    for (int r = 0; r < 8; ++r) {
        int m  = rowBase + r + 8 * hi;           // C/D layout: row = r (+8 hi half)
        int gi = idx[m];
        const float* x2row = x2 + (size_t)gi * D;
        #pragma unroll
        for (int nt = 0; nt < 8; ++nt) {
            int n = nt * 16 + nlo;
            float v = acc[nt][r] + bn[nt] + x2row[n];
            v = v > 0.f ? v : expm1f(v);         // ELU (alpha=1)
            hout[(size_t)m * D + n] = v;
        }
    }
}

// -------------------------------------------------------------------------
// Kernel 4: plain WMMA GEMM  out[:, colOff:colOff+128] = A @ W[:,slice] + b
// ACT: 0 = none, 1 = relu.
// -------------------------------------------------------------------------
template <int ACT>
__global__ __launch_bounds__(256)
void gemm_bias_kernel(const float* __restrict__ A,
                      const float* __restrict__ W, int ldw, int colOff,
                      const float* __restrict__ bias,
                      float* __restrict__ out, int ldo)
{
    __shared__ float WT[D * D];
    stage_WT(W, ldw, colOff, WT);

    int wave = threadIdx.x >> 5;
    int lane = threadIdx.x & 31;
    int nlo  = lane & 15;
    int hi   = lane >> 4;
    int rowBase = blockIdx.x * 128 + wave * 16;

    v8f acc[8] = {};
    gemm_acc_wave(A, rowBase, WT, acc);

    float bn[8];
    #pragma unroll
    for (int nt = 0; nt < 8; ++nt) bn[nt] = bias[colOff + nt * 16 + nlo];

    #pragma unroll
    for (int r = 0; r < 8; ++r) {
        int m = rowBase + r + 8 * hi;
        #pragma unroll
        for (int nt = 0; nt < 8; ++nt) {
            int n = nt * 16 + nlo;
            float v = acc[nt][r] + bn[nt];
            if (ACT == 1) v = fmaxf(v, 0.f);
            out[(size_t)m * ldo + colOff + n] = v;
        }
    }
}

// -------------------------------------------------------------------------
// Kernel 5: tiny head projection  out = y1 @ W_f2 + b_f2   (256 -> 10)
// -------------------------------------------------------------------------
__global__ void head2_kernel(const float* __restrict__ y1,
                             const float* __restrict__ W2,
                             const float* __restrict__ b2,
                             float* __restrict__ out)
{
    int gidx = blockIdx.x;
    int t = threadIdx.x;
    if (t >= NTASKS) return;
    const float* yrow = y1 + (size_t)gidx * (2 * D);
    float acc = b2[t];
    for (int k = 0; k < 2 * D; ++k) acc += yrow[k] * W2[k * NTASKS + t];
    out[(size_t)gidx * NTASKS + t] = acc;
}

// -------------------------------------------------------------------------
extern "C" void kernel_launch(void* const* d_in, const int* in_sizes, int n_in,
                              void* d_out, int out_size, void* d_ws, size_t ws_size,
                              hipStream_t stream)
{
    const float* h0   = (const float*)d_in[0];   // [NSUB, D]
    const int*   idx  = (const int*)  d_in[1];   // [NSUB], sorted
    const float* Wfc  = (const float*)d_in[2];   // [L, D, D]
    const float* bfc  = (const float*)d_in[3];   // [L, D]
    const float* Wsum = (const float*)d_in[4];   // [L, D, D]
    const float* bsum = (const float*)d_in[5];   // [L, D]
    const float* Wf1  = (const float*)d_in[6];   // [D, 2D]
    const float* bf1  = (const float*)d_in[7];   // [2D]
    const float* Wf2  = (const float*)d_in[8];   // [2D, NTASKS]
    const float* bf2  = (const float*)d_in[9];   // [NTASKS]
    float* out = (float*)d_out;

    // workspace partition
    char* p = (char*)d_ws;
    int*   seg_start = (int*)p;   p += (((G + 1) * sizeof(int)) + 255) & ~(size_t)255;
    float* gmean     = (float*)p; p += (size_t)G * D * sizeof(float);
    float* x2        = (float*)p; p += (size_t)G * D * sizeof(float);
    float* y1        = (float*)p; p += (size_t)G * 2 * D * sizeof(float);
    float* h         = (float*)p; // NSUB * D floats

    seg_start_kernel<<<(G + 1 + 255) / 256, 256, 0, stream>>>(idx, seg_start, NSUB, G);

    const float* hin = h0;
    for (int i = 0; i < LNUM; ++i) {
        seg_mean_kernel<<<G, D, 0, stream>>>(hin, seg_start, gmean);
        gemm_bias_kernel<0><<<G / 128, 256, 0, stream>>>(
            gmean, Wsum + (size_t)i * D * D, D, 0, bsum + (size_t)i * D, x2, D);
        layer_fused_kernel<<<NSUB / 128, 256, 0, stream>>>(
            hin, Wfc + (size_t)i * D * D, bfc + (size_t)i * D, x2, idx, h);
        hin = h;   // layers 2,3 run in place on workspace h
    }

    seg_mean_kernel<<<G, D, 0, stream>>>(hin, seg_start, gmean);
    // y1 = relu(gmean @ W_f1 + b_f1), 256 cols done as two 128-col slices
    gemm_bias_kernel<1><<<G / 128, 256, 0, stream>>>(gmean, Wf1, 2 * D, 0, bf1, y1, 2 * D);
    gemm_bias_kernel<1><<<G / 128, 256, 0, stream>>>(gmean, Wf1, 2 * D, D, bf1, y1, 2 * D);
    head2_kernel<<<G, 32, 0, stream>>>(y1, Wf2, bf2, out);
}